// HierarchicalGCN_65970697666821
// MI455X (gfx1250) — compile-verified
//
#include <hip/hip_runtime.h>

// ---------------- problem constants ----------------
#define NODES      128
#define GROUPS     16
#define GSIZE      8
#define DIM        64
#define NB         8192                    // batch of graphs
#define N1         (NB * NODES)            // 1048576
#define N2         (NB * GROUPS)           // 131072
#define E1         (N1 * 8)                // 8388608
#define E2         (N2 * 8)                // 1048576

typedef __attribute__((ext_vector_type(2))) float v2f;
typedef __attribute__((ext_vector_type(8))) float v8f;

#define LDPAD 4   // row pad (dwords): LDS bank = (4*m + c) % 64 -> conflict-free frags

// =====================================================================
// Generic fp32 WMMA GEMM:  C[r, 0:64] = A[r, 0:K] @ W[K, 64] (+ bias)
// block = 128 threads (4 waves); block tile = 64 rows x 64 cols.
// blockIdx.y selects a "group" (A += gA, W += gW, C += gC, bias += gB).
// =====================================================================
__global__ __launch_bounds__(128) void gemm_wmma_k(
    const float* __restrict__ A, long long rowStride,
    const float* __restrict__ W, int K,
    float* __restrict__ C, long long cStride,
    const float* __restrict__ bias,
    long long gA, long long gW, long long gC, long long gB,
    int rows)
{
    __shared__ float As[64][64 + LDPAD];
    __shared__ float Wt[64][64 + LDPAD];   // Wt[n][k] = W[k][n]

    A += (long long)blockIdx.y * gA;
    W += (long long)blockIdx.y * gW;
    C += (long long)blockIdx.y * gC;
    const float* bptr = bias ? (bias + (long long)blockIdx.y * gB) : nullptr;

    const int tid  = threadIdx.x;
    const int wave = tid >> 5;         // 0..3 -> 16-row slab
    const int lane = tid & 31;
    const int row0 = blockIdx.x * 64;
    if (row0 >= rows) return;

    const int m     = lane & 15;       // M (or N) index within tile
    const int khalf = (lane >> 4) << 1; // 0 or 2 : K sub-offset per ISA layout

    v8f acc0 = {}, acc1 = {}, acc2 = {}, acc3 = {};

    for (int kc = 0; kc < K; kc += 64) {
        // ---- cooperative loads: 64x64 chunk of A, 64x64 chunk of W (transposed) ----
        #pragma unroll
        for (int i = tid; i < 64 * 16; i += 128) {
            const int r  = i >> 4;
            const int c4 = (i & 15) << 2;
            const float4 v = *(const float4*)(A + (long long)(row0 + r) * rowStride + kc + c4);
            As[r][c4 + 0] = v.x; As[r][c4 + 1] = v.y;
            As[r][c4 + 2] = v.z; As[r][c4 + 3] = v.w;
        }
        #pragma unroll
        for (int i = tid; i < 64 * 16; i += 128) {
            const int k  = i >> 4;
            const int c4 = (i & 15) << 2;
            const float4 v = *(const float4*)(W + (long long)(kc + k) * 64 + c4);
            Wt[c4 + 0][k] = v.x; Wt[c4 + 1][k] = v.y;
            Wt[c4 + 2][k] = v.z; Wt[c4 + 3][k] = v.w;
        }
        if (kc + 64 < K) {  // prefetch next A chunk (global_prefetch_b8)
            __builtin_prefetch(A + (long long)(row0 + (tid & 63)) * rowStride + kc + 64, 0, 0);
        }
        __syncthreads();

        // ---- 16 k-steps of V_WMMA_F32_16X16X4_F32, A frag reused over 4 B tiles ----
        #pragma unroll
        for (int ks = 0; ks < 16; ++ks) {
            const int kk = (ks << 2) + khalf;
            v2f a = *(const v2f*)&As[wave * 16 + m][kk];
            v2f b0 = *(const v2f*)&Wt[ 0 + m][kk];
            v2f b1 = *(const v2f*)&Wt[16 + m][kk];
            v2f b2 = *(const v2f*)&Wt[32 + m][kk];
            v2f b3 = *(const v2f*)&Wt[48 + m][kk];
            acc0 = __builtin_amdgcn_wmma_f32_16x16x4_f32(false, a, false, b0, (short)0, acc0, false, false);
            acc1 = __builtin_amdgcn_wmma_f32_16x16x4_f32(false, a, false, b1, (short)0, acc1, false, false);
            acc2 = __builtin_amdgcn_wmma_f32_16x16x4_f32(false, a, false, b2, (short)0, acc2, false, false);
            acc3 = __builtin_amdgcn_wmma_f32_16x16x4_f32(false, a, false, b3, (short)0, acc3, false, false);
        }
        __syncthreads();
    }

    // ---- epilogue: lane L vgpr j -> row (L/16)*8 + j, col L%16 ----
    const int col   = lane & 15;
    const int rbase = row0 + wave * 16 + ((lane >> 4) << 3);
    v8f accs[4] = { acc0, acc1, acc2, acc3 };
    #pragma unroll
    for (int t = 0; t < 4; ++t) {
        const int n  = t * 16 + col;
        const float bv = bptr ? bptr[n] : 0.0f;
        #pragma unroll
        for (int j = 0; j < 8; ++j) {
            C[(long long)(rbase + j) * cStride + n] = accs[t][j] + bv;
        }
    }
}

// =====================================================================
// degree: deg[dst[e]] += 1  (self loop folded into rsqrt kernel as +1)
// =====================================================================
__global__ __launch_bounds__(256) void deg_k(const int* __restrict__ dst,
                                             float* __restrict__ deg, int E)
{
    const long long t = (long long)blockIdx.x * 256 + threadIdx.x;
    if (t < E) unsafeAtomicAdd(&deg[dst[t]], 1.0f);
}

__global__ __launch_bounds__(256) void rsqrt_k(float* __restrict__ d, int N)
{
    const long long t = (long long)blockIdx.x * 256 + threadIdx.x;
    if (t < N) d[t] = rsqrtf(d[t] + 1.0f);   // +1 self loop; always > 0
}

// =====================================================================
// edge aggregation: acc[dst] += h[src] * dinv[src]*dinv[dst]
// 16 threads per edge, each handles one float4 (coalesced 256B row reads)
// =====================================================================
__global__ __launch_bounds__(256) void agg_k(const float* __restrict__ h,
                                             const int* __restrict__ src,
                                             const int* __restrict__ dst,
                                             const float* __restrict__ dinv,
                                             float* __restrict__ acc, int E)
{
    const long long t = (long long)blockIdx.x * 256 + threadIdx.x;
    const long long e = t >> 4;
    if (e >= E) return;
    const int q = (int)(t & 15);
    const int s = src[e];
    const int d = dst[e];
    const float coef = dinv[s] * dinv[d];
    const float4 v = *(const float4*)(h + (long long)s * DIM + q * 4);
    float* p = acc + (long long)d * DIM + q * 4;
    unsafeAtomicAdd(p + 0, v.x * coef);
    unsafeAtomicAdd(p + 1, v.y * coef);
    unsafeAtomicAdd(p + 2, v.z * coef);
    unsafeAtomicAdd(p + 3, v.w * coef);
}

// =====================================================================
// finalize: acc[n,d] += h[n,d] * dinv[n]^2 (self loop) + bias[d]
// =====================================================================
__global__ __launch_bounds__(256) void fin_k(const float* __restrict__ h,
                                             const float* __restrict__ dinv,
                                             const float* __restrict__ bias,
                                             float* __restrict__ acc, long long N)
{
    const long long t = (long long)blockIdx.x * 256 + threadIdx.x;
    if (t >= N * DIM) return;
    const long long n = t >> 6;
    const int d = (int)(t & 63);
    const float di = dinv[n];
    acc[t] += h[t] * di * di + bias[d];
}

// =====================================================================
// host launcher
// =====================================================================
extern "C" void kernel_launch(void* const* d_in, const int* in_sizes, int n_in,
                              void* d_out, int out_size, void* d_ws, size_t ws_size,
                              hipStream_t stream)
{
    const float* x       = (const float*)d_in[0];
    const int*   A1      = (const int*)  d_in[1];   // [2, E1]
    const int*   A2      = (const int*)  d_in[2];   // [2, E2]
    const float* lower_W = (const float*)d_in[3];
    const float* lower_b = (const float*)d_in[4];
    const float* group_W = (const float*)d_in[5];   // [16, 512, 64]
    const float* group_b = (const float*)d_in[6];   // [16, 64]
    const float* upper_W = (const float*)d_in[7];
    const float* upper_b = (const float*)d_in[8];
    float* out = (float*)d_out;                     // [N2, 64]

    char* ws = (char*)d_ws;
    float* h     = (float*)(ws);                         // N1*64 f32 (256 MB)
    float* acc1  = (float*)(ws + 268435456LL);           // N1*64 f32 (256 MB)
    float* dinv1 = (float*)(ws + 536870912LL);           // N1 f32
    float* gout  = (float*)(ws + 541065216LL);           // N2*64 f32
    float* u     = (float*)(ws + 574619648LL);           // N2*64 f32
    float* dinv2 = (float*)(ws + 608174080LL);           // N2 f32

    // deterministic per-call init of accumulators / degrees / output
    hipMemsetAsync(acc1,  0, (size_t)N1 * DIM * sizeof(float), stream);
    hipMemsetAsync(dinv1, 0, (size_t)N1 * sizeof(float), stream);
    hipMemsetAsync(dinv2, 0, (size_t)N2 * sizeof(float), stream);
    hipMemsetAsync(out,   0, (size_t)N2 * DIM * sizeof(float), stream);

    // degrees + symmetric norm factors
    deg_k<<<E1 / 256, 256, 0, stream>>>(A1 + E1, dinv1, E1);
    deg_k<<<E2 / 256, 256, 0, stream>>>(A2 + E2, dinv2, E2);
    rsqrt_k<<<N1 / 256, 256, 0, stream>>>(dinv1, N1);
    rsqrt_k<<<N2 / 256, 256, 0, stream>>>(dinv2, N2);

    // ---- lower GCN ----
    gemm_wmma_k<<<dim3(N1 / 64, 1), 128, 0, stream>>>(
        x, 64, lower_W, 64, h, 64, nullptr, 0, 0, 0, 0, N1);
    agg_k<<<(E1 * 16) / 256, 256, 0, stream>>>(h, A1, A1 + E1, dinv1, acc1, E1);
    fin_k<<<(N1 * DIM) / 256, 256, 0, stream>>>(h, dinv1, lower_b, acc1, (long long)N1);

    // ---- per-group GEMM: rows = graphs (stride NODES*DIM), K = 512 ----
    gemm_wmma_k<<<dim3(NB / 64, GROUPS), 128, 0, stream>>>(
        acc1, (long long)NODES * DIM,            // A row = graph b, base + g*8*64
        group_W, GSIZE * DIM,                    // W_g : 512 x 64
        gout, (long long)GROUPS * DIM,           // C row = graph b, base + g*64
        group_b,
        (long long)GSIZE * DIM,                  // gA  = 512
        (long long)GSIZE * DIM * DIM,            // gW  = 32768
        (long long)DIM,                          // gC  = 64
        (long long)DIM,                          // gB  = 64
        NB);

    // ---- upper GCN ----
    gemm_wmma_k<<<dim3(N2 / 64, 1), 128, 0, stream>>>(
        gout, 64, upper_W, 64, u, 64, nullptr, 0, 0, 0, 0, N2);
    agg_k<<<(E2 * 16) / 256, 256, 0, stream>>>(u, A2, A2 + E2, dinv2, out, E2);
    fin_k<<<((long long)N2 * DIM) / 256, 256, 0, stream>>>(u, dinv2, upper_b, out, (long long)N2);
}